// GNNLSTMModel_29764123361839
// MI455X (gfx1250) — compile-verified
//
#include <hip/hip_runtime.h>
#include <math.h>

// ---------------------------------------------------------------------------
// GNN (GINEConv x2) + LSTM pipeline for MI455X (gfx1250), f32 end-to-end using
// V_WMMA_F32_16X16X4_F32 (wave32). Fragment layouts per CDNA5 ISA 7.12.2:
//   A (16x4 f32): lane L holds A[L%16][4*kk + 2*(L/16) + {0,1}]   (v2f)
//   B (4x16 f32): lane L holds B[2*(L/16) + {0,1}][L%16]          (v2f)
//   C/D (16x16):  VGPR v, lane L -> row v + 8*(L/16), col L%16    (v8f)
//
// Round-4 change: the gather->atomic epilogue in gine_message is split into a
// batched gather phase (8 outstanding global_load_b32, progressive waits) and
// a compute+atomic phase, instead of the serialized load/wait-0/atomic chain
// the compiler produced when it recycled one VGPR.  All other structure
// (LDS-staged weights at stride 68, load/wmma phase split, register-resident
// W_hh fragments across 200 LSTM steps, 227KB dynamic LDS state) retained.
// ---------------------------------------------------------------------------

typedef __attribute__((ext_vector_type(2))) float v2f;
typedef __attribute__((ext_vector_type(8))) float v8f;

#define N_NODES 50000
#define N_EDGES 800000
#define HID     64
#define LSTM_F  32
#define LSTM_H  128
#define BATCH   64
#define TSTEPS  200
#define DIN     (HID + LSTM_F)   // 96
#define G4H     (4 * LSTM_H)     // 512
#define LN_EPS  1e-5f
#define WST     68               // LDS row stride (floats) for 64-wide tiles
#define HS      132              // LDS row stride (floats) for 128-wide tiles

__device__ __forceinline__ v8f wmma4(v2f a, v2f b, v8f c) {
  return __builtin_amdgcn_wmma_f32_16x16x4_f32(false, a, false, b,
                                               (short)0, c, false, false);
}

__device__ __forceinline__ float sigmoidf_(float x) {
  return 1.0f / (1.0f + expf(-x));
}

__device__ __forceinline__ v2f ld2(const float* p) { return *(const v2f*)p; }

// ---------------------------------------------------------------------------
// Kernel 1: fused edge linear + gather + relu + scatter-add.
// linW staged in LDS once per block; one wave per 16-edge tile.
// ---------------------------------------------------------------------------
__global__ void gine_message_kernel(const float* __restrict__ edge_attr,
                                    const float* __restrict__ linW,  // [64,64]
                                    const float* __restrict__ linb,  // [64]
                                    const float* __restrict__ h,     // [N,64]
                                    const int*   __restrict__ src,
                                    const int*   __restrict__ dst,
                                    float* __restrict__ agg)         // [N,64] zeroed
{
  __shared__ float wlds[HID * WST];
  __shared__ float blds[HID];

  for (int i = threadIdx.x; i < HID * HID; i += blockDim.x)
    wlds[(i >> 6) * WST + (i & 63)] = linW[i];
  if (threadIdx.x < HID) blds[threadIdx.x] = linb[threadIdx.x];
  __syncthreads();

  const int lane  = threadIdx.x & 31;
  const int wave  = (blockIdx.x * blockDim.x + threadIdx.x) >> 5;
  const int ebase = wave * 16;                 // exact grid: N_EDGES%16==0
  if (ebase >= N_EDGES) return;
  const int m    = lane & 15;
  const int half = lane >> 4;
  const int klo  = half * 2;

  // A fragments: 16 edge rows, K=64 -> 16 k-steps (global, streamed)
  v2f afrag[16];
  const float* arow = edge_attr + (size_t)(ebase + m) * HID;
#pragma unroll
  for (int kk = 0; kk < 16; ++kk)
    afrag[kk] = ld2(arow + kk * 4 + klo);

  // Edge endpoints: lanes of one half all share e for a given v -> the
  // per-v gather/scatter below is 16 consecutive floats (coalesced).
  int se[8], de[8];
#pragma unroll
  for (int v = 0; v < 8; ++v) {
    const int e = ebase + v + 8 * half;
    se[v] = src[e];
    de[v] = dst[e];
    __builtin_assume(se[v] >= 0 && se[v] < N_NODES);
    __builtin_assume(de[v] >= 0 && de[v] < N_NODES);
  }

#pragma unroll
  for (int nt = 0; nt < 4; ++nt) {
    const int n = nt * 16 + m;
    const float bias = blds[n];
    const float* brow = &wlds[n * WST];        // B[k][n] = linW[n][k]
    v2f bf[16];
#pragma unroll
    for (int kk = 0; kk < 16; ++kk)            // load phase (ds burst)
      bf[kk] = ld2(brow + kk * 4 + klo);
    v8f acc;
#pragma unroll
    for (int i = 0; i < 8; ++i) acc[i] = bias;
#pragma unroll
    for (int kk = 0; kk < 16; ++kk)            // wmma phase
      acc = wmma4(afrag[kk], bf[kk], acc);

    // Epilogue phase 1: batched gather (8 outstanding loads, distinct regs)
    float hv[8];
#pragma unroll
    for (int v = 0; v < 8; ++v) {
      const int hoff = se[v] * HID + n;        // < 3.2M, fits 32-bit
      __builtin_assume(hoff >= 0);
      hv[v] = h[hoff];
    }
    // Epilogue phase 2: relu + atomic scatter
#pragma unroll
    for (int v = 0; v < 8; ++v) {
      const int aoff = de[v] * HID + n;
      __builtin_assume(aoff >= 0);
      const float val = fmaxf(acc[v] + hv[v], 0.0f);
      atomicAdd(&agg[aoff], val);
    }
  }
}

// ---------------------------------------------------------------------------
// Kernel 2: fused node update.  W1/W2 staged in LDS; 4 waves/block (128 thr),
// one wave per 16-node tile; per-wave 16x68 LDS tile for the t-transpose and
// the layernorm.
// ---------------------------------------------------------------------------
__global__ void gine_node_kernel(const float* __restrict__ h,
                                 const float* __restrict__ agg,
                                 const float* __restrict__ W1,
                                 const float* __restrict__ b1,
                                 const float* __restrict__ W2,
                                 const float* __restrict__ b2,
                                 const float* __restrict__ epsp,   // [1]
                                 const float* __restrict__ lnsc,   // [64]
                                 const float* __restrict__ lnbi,   // [64]
                                 float* __restrict__ hout)
{
  __shared__ float w1lds[HID * WST];
  __shared__ float w2lds[HID * WST];
  __shared__ float tile[4][16][WST];
  __shared__ float b1lds[HID], b2lds[HID], sclds[HID], bilds[HID];

  for (int i = threadIdx.x; i < HID * HID; i += blockDim.x) {
    const int r = i >> 6, c = i & 63;
    w1lds[r * WST + c] = W1[i];
    w2lds[r * WST + c] = W2[i];
  }
  if (threadIdx.x < HID) {
    b1lds[threadIdx.x] = b1[threadIdx.x];
    b2lds[threadIdx.x] = b2[threadIdx.x];
    sclds[threadIdx.x] = lnsc[threadIdx.x];
    bilds[threadIdx.x] = lnbi[threadIdx.x];
  }
  __syncthreads();

  const int lane  = threadIdx.x & 31;
  const int wslot = threadIdx.x >> 5;
  const int wave  = (blockIdx.x * blockDim.x + threadIdx.x) >> 5;
  int nbase = wave * 16;
  if (nbase > N_NODES - 16) nbase = N_NODES - 16;   // clamp: dup work, same data
  const int m    = lane & 15;
  const int half = lane >> 4;
  const int klo  = half * 2;
  float (*T)[WST] = tile[wslot];

  const float eps1 = 1.0f + epsp[0];

  // A fragments of z = (1+eps)*h + agg, computed during load
  v2f afrag[16];
  {
    const int roff = (nbase + m) * HID;        // < 3.2M
    __builtin_assume(roff >= 0);
    const float* hrow = h   + roff;
    const float* grow = agg + roff;
#pragma unroll
    for (int kk = 0; kk < 16; ++kk) {
      const v2f hv = ld2(hrow + kk * 4 + klo);
      const v2f gv = ld2(grow + kk * 4 + klo);
      afrag[kk].x = eps1 * hv.x + gv.x;
      afrag[kk].y = eps1 * hv.y + gv.y;
    }
  }

  // GEMM 1: relu(z @ W1^T + b1) -> LDS tile (row-major)
#pragma unroll
  for (int nt = 0; nt < 4; ++nt) {
    const int n = nt * 16 + m;
    const float* brow = &w1lds[n * WST];
    v2f bf[16];
#pragma unroll
    for (int kk = 0; kk < 16; ++kk)
      bf[kk] = ld2(brow + kk * 4 + klo);
    v8f acc;
    const float bias = b1lds[n];
#pragma unroll
    for (int i = 0; i < 8; ++i) acc[i] = bias;
#pragma unroll
    for (int kk = 0; kk < 16; ++kk)
      acc = wmma4(afrag[kk], bf[kk], acc);
#pragma unroll
    for (int v = 0; v < 8; ++v)
      T[v + 8 * half][n] = fmaxf(acc[v], 0.0f);
  }
  __syncthreads();

  // Reload t in A layout (padded stride -> conflict-free)
#pragma unroll
  for (int kk = 0; kk < 16; ++kk)
    afrag[kk] = ld2(&T[m][kk * 4 + klo]);
  __syncthreads();

  // GEMM 2: u = t @ W2^T + b2 (keep in regs, then dump to tile)
  v8f out[4];
#pragma unroll
  for (int nt = 0; nt < 4; ++nt) {
    const int n = nt * 16 + m;
    const float* brow = &w2lds[n * WST];
    v2f bf[16];
#pragma unroll
    for (int kk = 0; kk < 16; ++kk)
      bf[kk] = ld2(brow + kk * 4 + klo);
    v8f acc;
    const float bias = b2lds[n];
#pragma unroll
    for (int i = 0; i < 8; ++i) acc[i] = bias;
#pragma unroll
    for (int kk = 0; kk < 16; ++kk)
      acc = wmma4(afrag[kk], bf[kk], acc);
    out[nt] = acc;
  }
#pragma unroll
  for (int nt = 0; nt < 4; ++nt)
#pragma unroll
    for (int v = 0; v < 8; ++v)
      T[v + 8 * half][nt * 16 + m] = out[nt][v];
  __syncthreads();

  // Layernorm + relu: lanes 0..15 each own one node row
  if (half == 0) {
    const float* row = T[m];
    float s = 0.0f, s2 = 0.0f;
#pragma unroll
    for (int j = 0; j < HID; ++j) { const float x = row[j]; s += x; s2 += x * x; }
    const float mu   = s * (1.0f / HID);
    const float var  = s2 * (1.0f / HID) - mu * mu;
    const float rstd = rsqrtf(var + LN_EPS);
    const int ooff = (nbase + m) * HID;
    __builtin_assume(ooff >= 0);
    float* orow = hout + ooff;
#pragma unroll
    for (int j = 0; j < HID; ++j) {
      const float val = (row[j] - mu) * rstd * sclds[j] + bilds[j];
      orow[j] = fmaxf(val, 0.0f);
    }
  }
}

// ---------------------------------------------------------------------------
// Kernel 3: combined[t*B+b] = concat(hfin[nidx[b]], x_seq[b][t])
// ---------------------------------------------------------------------------
__global__ void build_combined_kernel(const float* __restrict__ hfin,
                                      const float* __restrict__ xseq,  // [B,T,F]
                                      const int*   __restrict__ nidx,  // [B]
                                      float* __restrict__ comb)        // [T*B,96]
{
  const int i = blockIdx.x * blockDim.x + threadIdx.x;
  if (i >= TSTEPS * BATCH * DIN) return;
  const int r = i / DIN, j = i % DIN;
  const int t = r / BATCH, b = r % BATCH;
  float v;
  if (j < HID) {
    const int nd = nidx[b];
    __builtin_assume(nd >= 0 && nd < N_NODES);
    v = hfin[nd * HID + j];
  } else {
    v = xseq[(b * TSTEPS + t) * LSTM_F + (j - HID)];
  }
  comb[i] = v;
}

// ---------------------------------------------------------------------------
// Kernel 4: x_proj = combined @ W_ih^T + b_ih    [12800,96]x[96,512]
// One wave per 16-row tile; B fragments register-staged per n-tile.
// ---------------------------------------------------------------------------
__global__ void xproj_kernel(const float* __restrict__ comb,
                             const float* __restrict__ Wih,   // [512,96]
                             const float* __restrict__ bih,   // [512]
                             float* __restrict__ xp)          // [T*B,512]
{
  const int lane = threadIdx.x & 31;
  const int wave = (blockIdx.x * blockDim.x + threadIdx.x) >> 5;
  const int rbase = wave * 16;
  if (rbase >= TSTEPS * BATCH) return;
  const int m = lane & 15, half = lane >> 4;
  const int klo = half * 2;

  v2f afrag[DIN / 4];
  const float* arow = comb + (rbase + m) * DIN;
#pragma unroll
  for (int kk = 0; kk < DIN / 4; ++kk)
    afrag[kk] = ld2(arow + kk * 4 + klo);

  for (int nt = 0; nt < G4H / 16; ++nt) {
    const int n = nt * 16 + m;
    const float* brow = Wih + n * DIN;
    v2f bf[DIN / 4];
#pragma unroll
    for (int kk = 0; kk < DIN / 4; ++kk)
      bf[kk] = ld2(brow + kk * 4 + klo);
    v8f acc;
    const float bias = bih[n];
#pragma unroll
    for (int i = 0; i < 8; ++i) acc[i] = bias;
#pragma unroll
    for (int kk = 0; kk < DIN / 4; ++kk)
      acc = wmma4(afrag[kk], bf[kk], acc);
#pragma unroll
    for (int v = 0; v < 8; ++v) {
      const int ooff = (rbase + v + 8 * half) * G4H + n;   // < 6.6M
      __builtin_assume(ooff >= 0);
      xp[ooff] = acc[v];
    }
  }
}

// ---------------------------------------------------------------------------
// Kernel 5: sequential LSTM (200 steps) + final layernorm + fc, single block.
// 1024 threads = 32 waves.  Wave w owns (gate g = w>>3, j-tile jt = w&7): its
// 32 W_hh B-fragments are registers, invariant across all steps/batch tiles.
// LDS: h0,h1,c at stride 132 (conflict-free A-frag reads) + gates[64][512].
// Total 227KB dynamic LDS (CDNA5 WGP: 320KB).
// ---------------------------------------------------------------------------
__global__ void lstm_seq_kernel(const float* __restrict__ xp,    // [T*B,512]
                                const float* __restrict__ Whh,   // [512,128]
                                const float* __restrict__ bhh,   // [512]
                                const float* __restrict__ lnsc,  // [128]
                                const float* __restrict__ lnbi,  // [128]
                                const float* __restrict__ fcW,   // [1,128]
                                const float* __restrict__ fcb,   // [1]
                                float* __restrict__ out)         // [64]
{
  extern __shared__ float smem[];
  float* hbuf0 = smem;                        // 64*HS
  float* hbuf1 = hbuf0 + BATCH * HS;          // 64*HS
  float* cbuf  = hbuf1 + BATCH * HS;          // 64*HS
  float* gates = cbuf  + BATCH * HS;          // 64*512

  for (int i = threadIdx.x; i < BATCH * HS; i += blockDim.x) {
    hbuf0[i] = 0.0f;
    cbuf[i]  = 0.0f;
  }
  __syncthreads();

  const int lane = threadIdx.x & 31;
  const int wv   = threadIdx.x >> 5;          // 0..31
  const int m    = lane & 15, half = lane >> 4;
  const int klo  = half * 2;

  const int g  = wv >> 3;                     // gate 0..3 (i,f,g,o)
  const int jt = wv & 7;                      // hidden tile 0..7
  const int nloc = jt * 16 + m;               // col within H
  const int n    = g * LSTM_H + nloc;         // col within 4H

  // Loop-invariant W_hh B-fragments: loaded ONCE for all 200 steps.
  v2f bfrag[LSTM_H / 4];
  {
    const float* wr = Whh + n * LSTM_H;       // B[k][n] = Whh[n][k]
#pragma unroll
    for (int kk = 0; kk < LSTM_H / 4; ++kk)
      bfrag[kk] = ld2(wr + kk * 4 + klo);
  }
  const float bb = bhh[n];

  float* hold = hbuf0;
  float* hnew = hbuf1;

  for (int t = 0; t < TSTEPS; ++t) {
    const int xoff_t = t * BATCH * G4H;       // < 6.6M, fits 32-bit
#pragma unroll
    for (int bt = 0; bt < 4; ++bt) {
      // A fragments from h_old (LDS, padded stride -> conflict-free)
      v2f af[LSTM_H / 4];
      const float* hr = hold + (bt * 16 + m) * HS;
#pragma unroll
      for (int kk = 0; kk < LSTM_H / 4; ++kk)
        af[kk] = ld2(hr + kk * 4 + klo);

      v8f acc;
#pragma unroll
      for (int v = 0; v < 8; ++v) {
        const int xo = xoff_t + (bt * 16 + v + 8 * half) * G4H + n;
        __builtin_assume(xo >= 0);
        acc[v] = xp[xo] + bb;
      }
#pragma unroll
      for (int kk = 0; kk < LSTM_H / 4; ++kk)
        acc = wmma4(af[kk], bfrag[kk], acc);
#pragma unroll
      for (int v = 0; v < 8; ++v)
        gates[(bt * 16 + v + 8 * half) * G4H + n] = acc[v];
    }
    __syncthreads();

    // Elementwise gate fusion + state update: 8192 cells / 1024 threads
    for (int e = threadIdx.x; e < BATCH * LSTM_H; e += blockDim.x) {
      const int b = e >> 7, j = e & (LSTM_H - 1);
      const float* gr = gates + b * G4H;
      const float i_ = sigmoidf_(gr[j]);
      const float f_ = sigmoidf_(gr[LSTM_H + j]);
      const float g_ = tanhf(gr[2 * LSTM_H + j]);
      const float o_ = sigmoidf_(gr[3 * LSTM_H + j]);
      const float c  = f_ * cbuf[b * HS + j] + i_ * g_;
      cbuf[b * HS + j] = c;
      hnew[b * HS + j] = o_ * tanhf(c);
    }
    __syncthreads();
    float* tmp = hold; hold = hnew; hnew = tmp;
  }

  // Final layernorm + fc: threads 0..63 each own one batch row
  if (threadIdx.x < BATCH) {
    const float* row = hold + threadIdx.x * HS;
    float s = 0.0f, s2 = 0.0f;
#pragma unroll
    for (int j = 0; j < LSTM_H; ++j) { const float x = row[j]; s += x; s2 += x * x; }
    const float mu   = s * (1.0f / LSTM_H);
    const float var  = s2 * (1.0f / LSTM_H) - mu * mu;
    const float rstd = rsqrtf(var + LN_EPS);
    float dot = 0.0f;
#pragma unroll
    for (int j = 0; j < LSTM_H; ++j)
      dot += ((row[j] - mu) * rstd * lnsc[j] + lnbi[j]) * fcW[j];
    out[threadIdx.x] = dot + fcb[0];
  }
}

// ---------------------------------------------------------------------------
// Host launch
// ---------------------------------------------------------------------------
extern "C" void kernel_launch(void* const* d_in, const int* in_sizes, int n_in,
                              void* d_out, int out_size, void* d_ws, size_t ws_size,
                              hipStream_t stream) {
  const float* x        = (const float*)d_in[0];
  const float* edgeattr = (const float*)d_in[1];
  const float* xseq     = (const float*)d_in[2];
  const float* linW     = (const float*)d_in[3];
  const float* linb     = (const float*)d_in[4];
  const float* mlpW1    = (const float*)d_in[5];
  const float* mlpb1    = (const float*)d_in[6];
  const float* mlpW2    = (const float*)d_in[7];
  const float* mlpb2    = (const float*)d_in[8];
  const float* eps      = (const float*)d_in[9];
  const float* lnsc     = (const float*)d_in[10];
  const float* lnbi     = (const float*)d_in[11];
  const float* Wih      = (const float*)d_in[12];
  const float* Whh      = (const float*)d_in[13];
  const float* bih      = (const float*)d_in[14];
  const float* bhh      = (const float*)d_in[15];
  const float* llnsc    = (const float*)d_in[16];
  const float* llnbi    = (const float*)d_in[17];
  const float* fcW      = (const float*)d_in[18];
  const float* fcb      = (const float*)d_in[19];
  const int*   eidx     = (const int*)d_in[20];
  const int*   nidx     = (const int*)d_in[21];
  const int* src = eidx;
  const int* dst = eidx + N_EDGES;

  // Workspace layout (floats)
  float* ws   = (float*)d_ws;
  float* h0   = ws;
  float* h1   = h0 + (size_t)N_NODES * HID;
  float* agg  = h1 + (size_t)N_NODES * HID;
  float* comb = agg + (size_t)N_NODES * HID;
  float* xpj  = comb + (size_t)TSTEPS * BATCH * DIN;

  const int msg_blocks  = (N_EDGES / 16) / 8;                 // 6250 (256 thr)
  const int node_blocks = ((N_NODES + 15) / 16 + 3) / 4;      // 782  (128 thr)
  const int comb_blocks = (TSTEPS * BATCH * DIN + 255) / 256; // 4800
  const int xp_blocks   = (TSTEPS * BATCH / 16) / 8;          // 100  (256 thr)

  // ---- GINEConv layer 0 (h_in = x) ----
  hipMemsetAsync(agg, 0, (size_t)N_NODES * HID * sizeof(float), stream);
  gine_message_kernel<<<msg_blocks, 256, 0, stream>>>(
      edgeattr, linW, linb, x, src, dst, agg);
  gine_node_kernel<<<node_blocks, 128, 0, stream>>>(
      x, agg, mlpW1, mlpb1, mlpW2, mlpb2, eps, lnsc, lnbi, h0);

  // ---- GINEConv layer 1 ----
  hipMemsetAsync(agg, 0, (size_t)N_NODES * HID * sizeof(float), stream);
  gine_message_kernel<<<msg_blocks, 256, 0, stream>>>(
      edgeattr, linW + HID * HID, linb + HID, h0, src, dst, agg);
  gine_node_kernel<<<node_blocks, 128, 0, stream>>>(
      h0, agg, mlpW1 + HID * HID, mlpb1 + HID, mlpW2 + HID * HID, mlpb2 + HID,
      eps + 1, lnsc + HID, lnbi + HID, h1);

  // ---- LSTM front-end ----
  build_combined_kernel<<<comb_blocks, 256, 0, stream>>>(h1, xseq, nidx, comb);
  xproj_kernel<<<xp_blocks, 256, 0, stream>>>(comb, Wih, bih, xpj);

  // ---- Sequential LSTM + LN + FC ----
  const int smem_bytes =
      (3 * BATCH * HS + BATCH * G4H) * (int)sizeof(float);   // 232,448 B
  (void)hipFuncSetAttribute((const void*)lstm_seq_kernel,
                            hipFuncAttributeMaxDynamicSharedMemorySize,
                            smem_bytes);
  lstm_seq_kernel<<<1, 1024, smem_bytes, stream>>>(
      xpj, Whh, bhh, llnsc, llnbi, fcW, fcb, (float*)d_out);
}